// PoleAgent001_37572373905493
// MI455X (gfx1250) — compile-verified
//
#include <hip/hip_runtime.h>

#define NY   64
#define NU   32
#define NH   512
#define NHRZ 2048
#define BB   128
#define NWG  8
#define SLICE 64          // NH / NWG columns of x_next owned per workgroup
#define WROW  544         // NH + NU, row length of xu2x_w

typedef __attribute__((ext_vector_type(2))) float v2f;
typedef __attribute__((ext_vector_type(8))) float v8f;

// fp32 tensor op: D(16x16) += A(16x4) * B(4x16)
__device__ __forceinline__ v8f wmma4(v2f a, v2f b, v8f c) {
  return __builtin_amdgcn_wmma_f32_16x16x4_f32(false, a, false, b, (short)0, c,
                                               false, false);
}

// ---------------------------------------------------------------------------
// Kernel 1: x0 = y0 @ y2x_w.T + y2x_b   (B x NY @ NY x NH), also resets barrier
// ---------------------------------------------------------------------------
__global__ __launch_bounds__(256) void pole_x0_kernel(
    const float* __restrict__ y0, const float* __restrict__ y2x_w,
    const float* __restrict__ y2x_b, float* __restrict__ xbuf0,
    unsigned* __restrict__ cnt) {
  if (blockIdx.x == 0 && threadIdx.x == 0) *cnt = 0u;  // reset grid barrier
  const int wv = threadIdx.x >> 5;      // wave id 0..7 -> 16-row batch strip
  const int l  = threadIdx.x & 31;
  const int lr = l & 15;                // row/col within tile
  const int kh = (l >> 4) << 1;         // K offset 0 or 2 (A/B operand layout)
  const int ph = l >> 4;
  const int colbase = blockIdx.x * SLICE;
  const int gr = wv * 16;

  v8f acc[4];
#pragma unroll
  for (int t = 0; t < 4; ++t) {
    float bb = y2x_b[colbase + t * 16 + lr];
#pragma unroll
    for (int i = 0; i < 8; ++i) acc[t][i] = bb;
  }
#pragma unroll
  for (int k0 = 0; k0 < NY; k0 += 4) {
    v2f a = *(const v2f*)(y0 + (size_t)(gr + lr) * NY + k0 + kh);
#pragma unroll
    for (int t = 0; t < 4; ++t) {
      v2f b = *(const v2f*)(y2x_w + (size_t)(colbase + t * 16 + lr) * NY + k0 + kh);
      acc[t] = wmma4(a, b, acc[t]);
    }
  }
#pragma unroll
  for (int t = 0; t < 4; ++t)
#pragma unroll
    for (int i = 0; i < 8; ++i)
      xbuf0[(size_t)(gr + i + (ph << 3)) * NH + colbase + t * 16 + lr] = acc[t][i];
}

// ---------------------------------------------------------------------------
// Kernel 2: persistent scan. 8 WGs, each owns a 512x64 slice of A in LDS.
// Per step: x_{k+1}[:,slice] = x_k @ A[:,slice] + u_k @ Bm[:,slice] + b,
// fused Y_k = x_k @ x2y_w.T + x2y_b (rows strip per WG), then grid barrier.
// ---------------------------------------------------------------------------
__global__ __launch_bounds__(256) void pole_scan_kernel(
    const float* __restrict__ U, const float* __restrict__ xu2x_w,
    const float* __restrict__ xu2x_b, const float* __restrict__ x2y_w,
    const float* __restrict__ x2y_b, float* __restrict__ Y, float* xbuf0,
    float* xbuf1, unsigned* cnt) {
  // LDS: A slice stored as K-pairs per output column: As2[n][p] = (A[2p][col],A[2p+1][col])
  //      = xu2x_w[col][2p..2p+1]; pair-row stride padded to 257 (2056B) -> conflict-free.
  __shared__ v2f As2[SLICE * 257];   // 131,584 B
  __shared__ v2f Bm2[SLICE * 17];    //   8,704 B

  const int tid = threadIdx.x;
  const int wv  = tid >> 5;
  const int l   = tid & 31;
  const int lr  = l & 15;
  const int kh  = (l >> 4) << 1;     // K offset 0/2 within a 4-wide WMMA step
  const int ph  = l >> 4;            // pair offset 0/1
  const int wg  = blockIdx.x;
  const int colbase = wg * SLICE;

  for (int idx = tid; idx < SLICE * 256; idx += 256) {
    int n = idx >> 8, p = idx & 255;
    As2[n * 257 + p] = *(const v2f*)(xu2x_w + (size_t)(colbase + n) * WROW + 2 * p);
  }
  for (int idx = tid; idx < SLICE * 16; idx += 256) {
    int n = idx >> 4, p = idx & 15;
    Bm2[n * 17 + p] = *(const v2f*)(xu2x_w + (size_t)(colbase + n) * WROW + NH + 2 * p);
  }
  float xb[4];
#pragma unroll
  for (int t = 0; t < 4; ++t) xb[t] = xu2x_b[colbase + t * 16 + lr];
  const float yb = x2y_b[(wv & 3) * 16 + lr];
  __syncthreads();

  const int gr = wv * 16;   // recurrence batch-row strip for this wave
  const int ry = wg * 16;   // Y batch-row strip for this WG

  for (int k = 0; k < NHRZ; ++k) {
    const float* xk = (k & 1) ? xbuf1 : xbuf0;
    float*       xn = (k & 1) ? xbuf0 : xbuf1;

    // ---- recurrence: 16x64 output per wave, K = 512 (x@A) + 32 (u@Bm) ----
    v8f acc[4];
#pragma unroll
    for (int t = 0; t < 4; ++t)
#pragma unroll
      for (int i = 0; i < 8; ++i) acc[t][i] = xb[t];

#pragma unroll 8
    for (int k0 = 0; k0 < NH; k0 += 4) {
      v2f a = *(const v2f*)(xk + (size_t)(gr + lr) * NH + k0 + kh);
#pragma unroll
      for (int t = 0; t < 4; ++t) {
        v2f b = As2[(t * 16 + lr) * 257 + (k0 >> 1) + ph];
        acc[t] = wmma4(a, b, acc[t]);
      }
    }
    const float* uk = U + (size_t)k * BB * NU;
#pragma unroll
    for (int k0 = 0; k0 < NU; k0 += 4) {
      v2f a = *(const v2f*)(uk + (size_t)(gr + lr) * NU + k0 + kh);
#pragma unroll
      for (int t = 0; t < 4; ++t) {
        v2f b = Bm2[(t * 16 + lr) * 17 + (k0 >> 1) + ph];
        acc[t] = wmma4(a, b, acc[t]);
      }
    }
#pragma unroll
    for (int t = 0; t < 4; ++t)
#pragma unroll
      for (int i = 0; i < 8; ++i)
        xn[(size_t)(gr + i + (ph << 3)) * NH + colbase + t * 16 + lr] = acc[t][i];

    // ---- fused output projection Y_k (waves 0..3, wave-uniform branch) ----
    if (wv < 4) {
      v8f ya;
#pragma unroll
      for (int i = 0; i < 8; ++i) ya[i] = yb;
#pragma unroll 8
      for (int k0 = 0; k0 < NH; k0 += 4) {
        v2f a = *(const v2f*)(xk + (size_t)(ry + lr) * NH + k0 + kh);
        v2f b = *(const v2f*)(x2y_w + (size_t)(wv * 16 + lr) * NH + k0 + kh);
        ya = wmma4(a, b, ya);
      }
#pragma unroll
      for (int i = 0; i < 8; ++i)
        Y[((size_t)k * BB + ry + i + (ph << 3)) * NY + wv * 16 + lr] = ya[i];
    }

    // ---- device-wide step barrier (monotonic counter, no reset needed) ----
    __builtin_amdgcn_fence(__ATOMIC_RELEASE, "agent");
    __syncthreads();
    if (tid == 0) {
      unsigned old = __hip_atomic_fetch_add(cnt, 1u, __ATOMIC_RELAXED,
                                            __HIP_MEMORY_SCOPE_AGENT);
      unsigned need = (old / (unsigned)NWG + 1u) * (unsigned)NWG;
      while (__hip_atomic_load(cnt, __ATOMIC_RELAXED,
                               __HIP_MEMORY_SCOPE_AGENT) < need)
        __builtin_amdgcn_s_sleep(2);
    }
    __syncthreads();
    __builtin_amdgcn_fence(__ATOMIC_ACQUIRE, "agent");
  }

  // ---- Y_{2048} from final state (in xbuf0 since 2048 is even) ----
  if (wv < 4) {
    v8f ya;
#pragma unroll
    for (int i = 0; i < 8; ++i) ya[i] = yb;
#pragma unroll 8
    for (int k0 = 0; k0 < NH; k0 += 4) {
      v2f a = *(const v2f*)(xbuf0 + (size_t)(ry + lr) * NH + k0 + kh);
      v2f b = *(const v2f*)(x2y_w + (size_t)(wv * 16 + lr) * NH + k0 + kh);
      ya = wmma4(a, b, ya);
    }
#pragma unroll
    for (int i = 0; i < 8; ++i)
      Y[((size_t)NHRZ * BB + ry + i + (ph << 3)) * NY + wv * 16 + lr] = ya[i];
  }
}

extern "C" void kernel_launch(void* const* d_in, const int* in_sizes, int n_in,
                              void* d_out, int out_size, void* d_ws,
                              size_t ws_size, hipStream_t stream) {
  (void)in_sizes; (void)n_in; (void)out_size; (void)ws_size;
  const float* y0     = (const float*)d_in[0];
  const float* U      = (const float*)d_in[1];
  const float* y2x_w  = (const float*)d_in[2];
  const float* y2x_b  = (const float*)d_in[3];
  const float* xu2x_w = (const float*)d_in[4];
  const float* xu2x_b = (const float*)d_in[5];
  const float* x2y_w  = (const float*)d_in[6];
  const float* x2y_b  = (const float*)d_in[7];
  float* Y = (float*)d_out;

  float*    xbuf0 = (float*)d_ws;                 // 128*512 f32 = 256 KB
  float*    xbuf1 = xbuf0 + (size_t)BB * NH;      // 256 KB
  unsigned* cnt   = (unsigned*)(xbuf1 + (size_t)BB * NH);

  pole_x0_kernel<<<NWG, 256, 0, stream>>>(y0, y2x_w, y2x_b, xbuf0, cnt);
  pole_scan_kernel<<<NWG, 256, 0, stream>>>(U, xu2x_w, xu2x_b, x2y_w, x2y_b, Y,
                                            xbuf0, xbuf1, cnt);
}